// T2I_OTGatedFusion_42511586296398
// MI455X (gfx1250) — compile-verified
//
#include <hip/hip_runtime.h>
#include <math.h>

// ---------------------------------------------------------------------------
// T2I OT-gated fusion for MI455X (gfx1250), wave32, WMMA f32 16x16x4 + TDM.
// Pipeline:
//   prep   : transpose text -> At, row norms nA/nB, init sinkhorn state
//   cost   : WMMA GEMM  E[n,m] = exp(-10 * cdist(A_n, B_m))   (134MB, L2-resident)
//   50x    : prep -> rows-matvec -> cols-matvec -> update  (faithful scan logic)
//   aligned: WMMA GEMM (recompute cost tile -> P=exp(-0.1*cost) in LDS,
//            then X(=imf*v) @ P^T, atomicAdd partials)        -> raw[128,256]
//   interp : t_line = lerp_N->W( u[n]*raw )                   -> [2,64,64]
//   fusion : TDM-staged tline + WMMA MLP, gated blend         -> out
// ---------------------------------------------------------------------------

typedef __attribute__((ext_vector_type(2))) float v2f;
typedef __attribute__((ext_vector_type(8))) float v8f;
typedef __attribute__((ext_vector_type(4))) unsigned int u32x4;
typedef __attribute__((ext_vector_type(8))) int i32x8;
typedef __attribute__((ext_vector_type(4))) int i32x4;

#define N_TXT 256
#define L_DIM 128
#define M_IMG 131072
#define C_CH  64
#define B_BT  2
#define W_OUT 64

static __device__ __forceinline__ v8f wmma4(v2f a, v2f b, v8f c) {
  // D = A(16x4 f32) x B(4x16 f32) + C(16x16 f32)
  return __builtin_amdgcn_wmma_f32_16x16x4_f32(
      /*neg_a=*/false, a, /*neg_b=*/false, b,
      /*c_mod=*/(short)0, c, /*reuse_a=*/false, /*reuse_b=*/false);
}

// TDM: DMA a contiguous run of `nelem` f32 from global -> LDS (2D tile n x 1).
// D# per CDNA5 ISA ch.8: group0 = {count/lds_addr/global_addr/type},
// group1 = {mask, data_size, dims, strides}; groups 2/3 zero (<=2D tensor).
// This toolchain exposes the 6-arg clang-23 builtin form.
static __device__ __forceinline__ void tdm_load_f32(unsigned lds_off,
                                                    const void* gptr,
                                                    unsigned nelem) {
  unsigned long long ga = (unsigned long long)(uintptr_t)gptr;
  u32x4 g0;
  g0[0] = 1u;                                        // count=1, no gather
  g0[1] = lds_off;                                   // lds_addr (bytes)
  g0[2] = (unsigned)(ga & 0xffffffffu);              // global_addr[31:0]
  g0[3] = (unsigned)((ga >> 32) & 0x01ffffffu)       // global_addr[56:32]
          | (2u << 30);                              // type=2 ("image")
  i32x8 g1;
  g1[0] = (int)(2u << 16);                           // data_size=2 -> 4 bytes
  g1[1] = (int)((nelem & 0xffffu) << 16);            // tensor_dim0[15:0]
  g1[2] = (int)((nelem >> 16) & 0xffffu)             // tensor_dim0[31:16]
          | (1 << 16);                               // tensor_dim1 = 1
  g1[3] = (int)((nelem & 0xffffu) << 16);            // tile_dim0 = nelem
  g1[4] = 1;                                         // tile_dim1=1, tile_dim2=0
  g1[5] = (int)nelem;                                // tensor_dim0_stride[31:0]
  g1[6] = 0;                                         // stride hi / dim1_stride
  g1[7] = 0;
  i32x4 z4 = {0, 0, 0, 0};
  i32x8 z8 = {0, 0, 0, 0, 0, 0, 0, 0};
  __builtin_amdgcn_tensor_load_to_lds(g0, g1, z4, z4, z8, 0);
  __builtin_amdgcn_s_wait_tensorcnt(0);
}

// ---------------------------------------------------------------------------
__global__ __launch_bounds__(256) void prep_kernel(
    const float* __restrict__ text, const float* __restrict__ image,
    float* At, float* nA, float* nB,
    float* u, float* v, float* pu, float* pv,
    float* raw, float* scal) {
  int idx = blockIdx.x * blockDim.x + threadIdx.x;
  if (idx < N_TXT * L_DIM) {
    int n = idx / L_DIM, l = idx % L_DIM;
    At[idx] = text[l * N_TXT + n];   // At[n][l] = tf[l][n]
    raw[idx] = 0.0f;
  }
  if (idx < N_TXT) {
    float s = 0.0f;
    for (int l = 0; l < L_DIM; ++l) { float t = text[l * N_TXT + idx]; s += t * t; }
    nA[idx] = s;
    u[idx] = 1.0f; pu[idx] = 1.0f;
  }
  if (idx < M_IMG) {
    float s = 0.0f;
    for (int l = 0; l < L_DIM; ++l) { float t = image[(size_t)l * M_IMG + idx]; s += t * t; }
    nB[idx] = s;
    v[idx] = 1.0f; pv[idx] = 1.0f;
  }
  if (idx == 0) { scal[0] = 0.0f; scal[1] = 0.0f; scal[2] = 0.0f; }
}

// ---------------------------------------------------------------------------
// E[n,m] = exp(-10*sqrt(max(nA[n]+nB[m]-2*dot,0)))
// Each block: 16-wide m-strip, each of 4 waves covers 4 n-tiles (64 rows).
__global__ __launch_bounds__(128) void cost_kernel(
    const float* __restrict__ At, const float* __restrict__ imf,
    const float* __restrict__ nA, const float* __restrict__ nB,
    float* __restrict__ E) {
  const int m0 = blockIdx.x * 16;
  const int wv = threadIdx.x >> 5, lane = threadIdx.x & 31;
  const int col = lane & 15;
  const int khalf = (lane < 16) ? 0 : 2;
  const int rhalf = (lane < 16) ? 0 : 8;
  const v8f zero = {0,0,0,0,0,0,0,0};
  v8f acc[4] = {zero, zero, zero, zero};
  for (int kk = 0; kk < L_DIM; kk += 4) {
    v2f b;  // B[k][j] = imf[kk+k][m0+j]
    b.x = imf[(size_t)(kk + khalf) * M_IMG + m0 + col];
    b.y = imf[(size_t)(kk + khalf + 1) * M_IMG + m0 + col];
#pragma unroll
    for (int t = 0; t < 4; ++t) {
      int n0 = wv * 64 + t * 16;
      v2f a = *(const v2f*)&At[(n0 + col) * L_DIM + kk + khalf];
      acc[t] = wmma4(a, b, acc[t]);
    }
  }
#pragma unroll
  for (int t = 0; t < 4; ++t) {
    int n0 = wv * 64 + t * 16;
#pragma unroll
    for (int i = 0; i < 8; ++i) {
      int n = n0 + i + rhalf;
      int m = m0 + col;
      float d2 = nA[n] + nB[m] - 2.0f * acc[t][i];
      float c = sqrtf(fmaxf(d2, 0.0f));
      E[(size_t)n * M_IMG + m] = __expf(-10.0f * c);   // exp(-cost/eps)
    }
  }
}

// ---------------------------------------------------------------------------
__global__ void sink_prep(float* scal, int first) {
  if (threadIdx.x == 0 && blockIdx.x == 0) {
    if (!first) {
      if (scal[0] < 1e-4f && scal[1] < 1e-4f) scal[2] = 1.0f;  // done |= conv
    }
    scal[0] = 0.0f; scal[1] = 0.0f;
  }
}

// nu[n] = 1/(E[n,:]@v + 100);  scal[0] += (nu-pu)^2
__global__ __launch_bounds__(256) void sink_rows(
    const float* __restrict__ E, const float* __restrict__ v,
    const float* __restrict__ pu, float* __restrict__ nu, float* scal) {
  __shared__ float red[256];
  int n = blockIdx.x, tid = threadIdx.x;
  const float* row = E + (size_t)n * M_IMG;
  float s = 0.0f;
  for (int m = tid; m < M_IMG; m += 256) {
    __builtin_prefetch(&row[m + 8 * 256], 0, 1);   // global_prefetch_b8
    s += row[m] * v[m];
  }
  red[tid] = s; __syncthreads();
  for (int off = 128; off > 0; off >>= 1) {
    if (tid < off) red[tid] += red[tid + off];
    __syncthreads();
  }
  if (tid == 0) {
    float val = 1.0f / (red[0] + 100.0f);
    nu[n] = val;
    float d = val - pu[n];
    atomicAdd(&scal[0], d * d);
  }
}

// nv[m] = 1/(E[:,m]@nu + 100);  scal[1] += (nv-pv)^2
__global__ __launch_bounds__(256) void sink_cols(
    const float* __restrict__ E, const float* __restrict__ nu,
    const float* __restrict__ pv, float* __restrict__ nv, float* scal) {
  __shared__ float snu[256];
  __shared__ float red[256];
  int tid = threadIdx.x;
  int m = blockIdx.x * 256 + tid;
  snu[tid] = nu[tid];
  __syncthreads();
  float s = 0.0f;
  for (int n = 0; n < N_TXT; ++n) {
    __builtin_prefetch(&E[(size_t)(n + 8) * M_IMG + m], 0, 1);
    s += E[(size_t)n * M_IMG + m] * snu[n];
  }
  float val = 1.0f / (s + 100.0f);
  nv[m] = val;
  float d = val - pv[m];
  red[tid] = d * d; __syncthreads();
  for (int off = 128; off > 0; off >>= 1) {
    if (tid < off) red[tid] += red[tid + off];
    __syncthreads();
  }
  if (tid == 0) atomicAdd(&scal[1], red[0]);
}

__global__ __launch_bounds__(256) void sink_update(
    const float* __restrict__ scal,
    const float* __restrict__ nu, const float* __restrict__ nv,
    float* u, float* v, float* pu, float* pv) {
  int idx = blockIdx.x * blockDim.x + threadIdx.x;
  bool done = scal[2] != 0.0f;
  bool conv = (scal[0] < 1e-4f) && (scal[1] < 1e-4f);
  if (idx < N_TXT) {
    if (!done) u[idx] = nu[idx];
    if (!(done || conv)) pu[idx] = nu[idx];
  }
  if (idx < M_IMG) {
    if (!done) v[idx] = nv[idx];
    if (!(done || conv)) pv[idx] = nv[idx];
  }
}

// ---------------------------------------------------------------------------
// raw[l,n] += sum_m imf[l,m]*v[m]*exp(-0.1*cost[n,m])   (u[n] applied later)
// Cost tile is RECOMPUTED here via WMMA (E underflows; exp(-0.1c) does not).
__global__ __launch_bounds__(128) void aligned_kernel(
    const float* __restrict__ At, const float* __restrict__ imf,
    const float* __restrict__ nA, const float* __restrict__ nB,
    const float* __restrict__ v, float* __restrict__ raw) {
  __shared__ __align__(16) float lds_P[16 * 68];
  __shared__ float lds_v[64];
  const int ntile = blockIdx.x % 16;
  const int lg = (blockIdx.x / 16) % 2;
  const int ms = blockIdx.x / 32;                 // 64-way m-split
  const int wv = threadIdx.x >> 5, lane = threadIdx.x & 31;
  const int col = lane & 15;
  const int khalf = (lane < 16) ? 0 : 2;
  const int rhalf = (lane < 16) ? 0 : 8;
  const int n0 = ntile * 16;
  const int l0 = lg * 64 + wv * 16;
  const v8f zero = {0,0,0,0,0,0,0,0};
  v8f acc = zero;
  for (int mc = ms * 2048; mc < ms * 2048 + 2048; mc += 64) {
    // phase 1: each wave recomputes one 16x16 cost sub-tile -> P in LDS
    v8f cacc = zero;
    const int msub = mc + wv * 16;
    for (int kk = 0; kk < L_DIM; kk += 4) {
      v2f bfr;
      bfr.x = imf[(size_t)(kk + khalf) * M_IMG + msub + col];
      bfr.y = imf[(size_t)(kk + khalf + 1) * M_IMG + msub + col];
      v2f afr = *(const v2f*)&At[(n0 + col) * L_DIM + kk + khalf];
      cacc = wmma4(afr, bfr, cacc);
    }
    __syncthreads();   // previous chunk's lds_P reads complete
#pragma unroll
    for (int i = 0; i < 8; ++i) {
      int n = n0 + i + rhalf;
      int m = msub + col;
      float d2 = nA[n] + nB[m] - 2.0f * cacc[i];
      float c = sqrtf(fmaxf(d2, 0.0f));
      lds_P[(i + rhalf) * 68 + wv * 16 + col] = __expf(-0.1f * c);  // exp(-cost*eps)
    }
    if (threadIdx.x < 64) lds_v[threadIdx.x] = v[mc + threadIdx.x];
    __syncthreads();
    // phase 2: acc += (imf[l-tile]*v) @ P^T over this 64-wide K chunk
    for (int kkl = 0; kkl < 64; kkl += 4) {
      v2f afr = *(const v2f*)&imf[(size_t)(l0 + col) * M_IMG + mc + kkl + khalf];
      afr.x *= lds_v[kkl + khalf];
      afr.y *= lds_v[kkl + khalf + 1];
      v2f bfr = *(const v2f*)&lds_P[col * 68 + kkl + khalf];
      acc = wmma4(afr, bfr, acc);
    }
  }
#pragma unroll
  for (int i = 0; i < 8; ++i) {
    int l = l0 + i + rhalf;
    int n = n0 + col;
    atomicAdd(&raw[l * N_TXT + n], acc[i]);
  }
}

// ---------------------------------------------------------------------------
// t_line[b,c,w] = lerp over n of u[n]*raw[l,n]  (trilinear, align_corners=False)
__global__ void interp_kernel(const float* __restrict__ raw,
                              const float* __restrict__ u,
                              float* __restrict__ tline) {
  int idx = blockIdx.x * blockDim.x + threadIdx.x;
  if (idx >= B_BT * C_CH * W_OUT) return;
  int w = idx % W_OUT;
  int lc = idx / W_OUT;
  float scale = (float)N_TXT / (float)W_OUT;
  float x = fmaxf(((float)w + 0.5f) * scale - 0.5f, 0.0f);
  int x0 = (int)floorf(x);
  if (x0 > N_TXT - 1) x0 = N_TXT - 1;
  int x1 = (x0 + 1 < N_TXT) ? x0 + 1 : N_TXT - 1;
  float f = x - (float)x0;
  float a0 = u[x0] * raw[lc * N_TXT + x0];
  float a1 = u[x1] * raw[lc * N_TXT + x1];
  tline[idx] = a0 * (1.0f - f) + a1 * f;
}

// ---------------------------------------------------------------------------
// Per-voxel MLP as WMMA GEMM: block = 64 voxels of one batch, 4 waves x 16 cols.
// tline slab (16KB) staged via Tensor Data Mover (TENSORcnt path).
__global__ __launch_bounds__(128) void fusion_kernel(
    const float* __restrict__ image, const float* __restrict__ tline,
    const float* __restrict__ w1, const float* __restrict__ b1,
    const float* __restrict__ w2, const float* __restrict__ b2,
    float* __restrict__ out) {
  __shared__ __align__(16) float lds_t[C_CH * W_OUT];   // 16 KB: tline[b][c][w]
  __shared__ __align__(16) float lds_h[4 * 16 * 68];    // per-wave h as [p][k]
  const int blocksPerBatch = M_IMG / 64;
  const int b = blockIdx.x / blocksPerBatch;
  const int p0blk = (blockIdx.x % blocksPerBatch) * 64;
  const int wv = threadIdx.x >> 5, lane = threadIdx.x & 31;
  const int col = lane & 15;
  const int khalf = (lane < 16) ? 0 : 2;
  const int rhalf = (lane < 16) ? 0 : 8;
  const int p0 = p0blk + wv * 16;
  const int wc = wv * 16 + col;                         // w coord of this column
  const float* imgb = image + (size_t)b * C_CH * M_IMG;

  // Wave 0 DMAs the whole tline[b] slab into LDS via the TDM, waits on
  // TENSORcnt, then the workgroup barriers before consuming it.
  if (threadIdx.x < 32) {
    unsigned lds_off = (unsigned)(uintptr_t)(&lds_t[0]);  // addr[31:0] == LDS offset
    tdm_load_f32(lds_off, &tline[b * C_CH * W_OUT], C_CH * W_OUT);
  }
  __syncthreads();

  const v8f zero = {0,0,0,0,0,0,0,0};
  // GEMM1: h = relu(W1[64x128] @ fused[128x16] + b1)
  v8f h[4] = {zero, zero, zero, zero};
  for (int kk = 0; kk < 128; kk += 4) {
    int k0 = kk + khalf;
    float f0 = (k0 < 64) ? lds_t[k0 * 64 + wc]
                         : imgb[(size_t)(k0 - 64) * M_IMG + p0 + col];
    float f1 = (k0 + 1 < 64) ? lds_t[(k0 + 1) * 64 + wc]
                             : imgb[(size_t)(k0 + 1 - 64) * M_IMG + p0 + col];
    v2f bf; bf.x = f0; bf.y = f1;
#pragma unroll
    for (int t = 0; t < 4; ++t) {
      v2f a = *(const v2f*)&w1[(t * 16 + col) * 128 + k0];
      h[t] = wmma4(a, bf, h[t]);
    }
  }
  // bias + relu, stage h into LDS as [p][k]
#pragma unroll
  for (int t = 0; t < 4; ++t)
#pragma unroll
    for (int i = 0; i < 8; ++i) {
      int ch = t * 16 + i + rhalf;
      lds_h[wv * (16 * 68) + col * 68 + ch] = fmaxf(h[t][i] + b1[ch], 0.0f);
    }
  __syncthreads();
  // GEMM2: g = W2[64x64] @ h[64x16] + b2
  v8f g[4] = {zero, zero, zero, zero};
  for (int kk = 0; kk < 64; kk += 4) {
    v2f bf = *(const v2f*)&lds_h[wv * (16 * 68) + col * 68 + kk + khalf];
#pragma unroll
    for (int t = 0; t < 4; ++t) {
      v2f a = *(const v2f*)&w2[(t * 16 + col) * 64 + kk + khalf];
      g[t] = wmma4(a, bf, g[t]);
    }
  }
  // gated blend epilogue
#pragma unroll
  for (int t = 0; t < 4; ++t)
#pragma unroll
    for (int i = 0; i < 8; ++i) {
      int ch = t * 16 + i + rhalf;
      float gate = 1.0f / (1.0f + __expf(-(g[t][i] + b2[ch])));
      float tv = lds_t[ch * 64 + wc];
      size_t idx = (size_t)(b * C_CH + ch) * M_IMG + p0 + col;
      float iv = imgb[(size_t)ch * M_IMG + p0 + col];
      out[idx] = iv * (1.0f - gate) + tv * gate;
    }
}

// ---------------------------------------------------------------------------
extern "C" void kernel_launch(void* const* d_in, const int* in_sizes, int n_in,
                              void* d_out, int out_size, void* d_ws, size_t ws_size,
                              hipStream_t stream) {
  (void)in_sizes; (void)n_in; (void)out_size; (void)ws_size;
  const float* text  = (const float*)d_in[0];
  const float* image = (const float*)d_in[1];   // == imf[128][M] flat
  const float* w1 = (const float*)d_in[2];
  const float* b1 = (const float*)d_in[3];
  const float* w2 = (const float*)d_in[4];
  const float* b2 = (const float*)d_in[5];
  float* out = (float*)d_out;

  float* ws = (float*)d_ws;
  size_t off = 0;
  float* E    = ws + off; off += (size_t)N_TXT * M_IMG;   // 134 MB
  float* At   = ws + off; off += N_TXT * L_DIM;
  float* nA   = ws + off; off += N_TXT;
  float* nB   = ws + off; off += M_IMG;
  float* u    = ws + off; off += N_TXT;
  float* v    = ws + off; off += M_IMG;
  float* pu   = ws + off; off += N_TXT;
  float* pv   = ws + off; off += M_IMG;
  float* nu   = ws + off; off += N_TXT;
  float* nv   = ws + off; off += M_IMG;
  float* raw  = ws + off; off += N_TXT * L_DIM;
  float* tln  = ws + off; off += B_BT * C_CH * W_OUT;
  float* scal = ws + off; off += 8;

  prep_kernel<<<512, 256, 0, stream>>>(text, image, At, nA, nB, u, v, pu, pv, raw, scal);
  cost_kernel<<<M_IMG / 16, 128, 0, stream>>>(At, image, nA, nB, E);
  for (int it = 0; it < 50; ++it) {
    sink_prep<<<1, 32, 0, stream>>>(scal, it == 0 ? 1 : 0);
    sink_rows<<<N_TXT, 256, 0, stream>>>(E, v, pu, nu, scal);
    sink_cols<<<M_IMG / 256, 256, 0, stream>>>(E, nu, pv, nv, scal);
    sink_update<<<M_IMG / 256, 256, 0, stream>>>(scal, nu, nv, u, v, pu, pv);
  }
  aligned_kernel<<<16 * 2 * 64, 128, 0, stream>>>(At, image, nA, nB, v, raw);
  interp_kernel<<<(B_BT * C_CH * W_OUT + 255) / 256, 256, 0, stream>>>(raw, u, tln);
  fusion_kernel<<<B_BT * (M_IMG / 64), 128, 0, stream>>>(image, tln, w1, b1, w2, b2, out);
}